// LossFunction_90366111907979
// MI455X (gfx1250) — compile-verified
//
#include <hip/hip_runtime.h>
#include <hip/hip_bf16.h>

// ---------------------------------------------------------------------------
// GE2E loss for MI455X (gfx1250, wave32).
//   phase1: normalize utterances, centroids, exclusive-centroid cosines.
//   phase2: fused  logits = 30 * xn @ cent_n^T  via V_WMMA_F32_16X16X4_F32,
//           B staged into LDS with global_load_async_to_lds_b128 (ASYNCcnt
//           double-buffered against the WMMA stream), diagonal replacement,
//           clamp, online sum-of-exp softmax, loss.
// Logits are bounded in [1e-6, 30] so exp() cannot overflow f32 -> no
// max-subtraction pass needed; single sweep over the 2048 columns.
// ---------------------------------------------------------------------------

typedef __attribute__((ext_vector_type(2))) float v2f;
typedef __attribute__((ext_vector_type(8))) float v8f;

#define NSPK   2048
#define NUTT   8
#define NDIM   128
#define NROWS  (NSPK * NUTT)   // 16384
#define LSCALE 30.0f

__device__ __forceinline__ float wave_sum32(float v) {
#pragma unroll
  for (int off = 16; off >= 1; off >>= 1)
    v += __shfl_xor(v, off, 32);
  return v;
}

// ---------------------------------------------------------------------------
// Phase 1: one block per speaker (256 threads = 8 waves, one wave per utt).
// Writes xn [16384x128], cent_n [2048x128], cos_exc [16384].
// ---------------------------------------------------------------------------
__global__ __launch_bounds__(256) void ge2e_phase1(
    const float* __restrict__ x, float* __restrict__ xn,
    float* __restrict__ centn, float* __restrict__ cosexc) {
  __shared__ float4 xs[NUTT][32];

  const int s    = blockIdx.x;
  const int w    = threadIdx.x >> 5;   // utterance index g
  const int lane = threadIdx.x & 31;

  const float4* row = (const float4*)(x + ((size_t)s * NUTT + w) * NDIM);
  float4 v = row[lane];

  float ss = v.x * v.x + v.y * v.y + v.z * v.z + v.w * v.w;
  ss = wave_sum32(ss);
  const float inv = 1.0f / fmaxf(sqrtf(ss), 1e-12f);
  v.x *= inv; v.y *= inv; v.z *= inv; v.w *= inv;

  ((float4*)(xn + ((size_t)s * NUTT + w) * NDIM))[lane] = v;
  xs[w][lane] = v;
  __syncthreads();

  // column sum over the 8 utterances of this speaker
  float4 cs = xs[0][lane];
#pragma unroll
  for (int g = 1; g < NUTT; ++g) {
    float4 t = xs[g][lane];
    cs.x += t.x; cs.y += t.y; cs.z += t.z; cs.w += t.w;
  }

  // exclusive centroid for utterance w:  (sum - xn_w) / (G-1)
  const float r1 = 1.0f / (float)(NUTT - 1);
  float4 e;
  e.x = (cs.x - v.x) * r1; e.y = (cs.y - v.y) * r1;
  e.z = (cs.z - v.z) * r1; e.w = (cs.w - v.w) * r1;
  float de = e.x * e.x + e.y * e.y + e.z * e.z + e.w * e.w;
  float dd = v.x * e.x + v.y * e.y + v.z * e.z + v.w * e.w;
  de = wave_sum32(de);
  dd = wave_sum32(dd);
  const float cexc = dd / fmaxf(sqrtf(de), 1e-8f);
  if (lane == 0) cosexc[s * NUTT + w] = cexc;

  // normalized centroid (wave 0 only)
  if (w == 0) {
    float4 c;
    c.x = cs.x * 0.125f; c.y = cs.y * 0.125f;
    c.z = cs.z * 0.125f; c.w = cs.w * 0.125f;
    float cn = c.x * c.x + c.y * c.y + c.z * c.z + c.w * c.w;
    cn = wave_sum32(cn);
    const float ci = 1.0f / fmaxf(sqrtf(cn), 1e-8f);
    c.x *= ci; c.y *= ci; c.z *= ci; c.w *= ci;
    ((float4*)(centn + (size_t)s * NDIM))[lane] = c;
  }
}

// ---------------------------------------------------------------------------
// Phase 2: one block per 16-row tile (8 waves). Each wave sweeps 16 N-tiles
// of 16 columns; B chunks (16 cols x 32 dims = 2 KB) are async-copied
// global -> LDS (4x global_load_async_to_lds_b128, 512 B each) one chunk
// ahead of the matrix unit, so WMMA issue overlaps the copies (ASYNCcnt).
//
// f32 16x16x4 WMMA fragment layout (ISA 7.12.2):
//   A (16x4):  v0 = A[m][k0 + 2*hi], v1 = A[m][k0 + 2*hi + 1],  m = m0 + lane%16
//   B (4x16):  v0 = B[k0 + 2*hi][n], v1 = B[k0+2*hi+1][n],      n = n0 + lane%16
//   C/D (16x16): VGPR r holds row m0+r (lanes 0-15) / m0+r+8 (lanes 16-31),
//                column = n0 + lane%16.
// ---------------------------------------------------------------------------
__global__ __launch_bounds__(256) void ge2e_phase2(
    const float* __restrict__ xn, const float* __restrict__ centn,
    const float* __restrict__ cosexc, float* __restrict__ out) {
  // Per-wave double buffer: [buf][wave][16 cols x 32 dims] = 32 KB total.
  __shared__ float bstage[2][8][16 * 32];
  __shared__ float ce_s[16];
  __shared__ float part[8][16];

  const int tid  = threadIdx.x;
  const int w    = tid >> 5;
  const int lane = tid & 31;
  const int lm   = lane & 15;
  const int hi   = lane >> 4;
  const int m0   = blockIdx.x * 16;

  if (tid < 16) ce_s[tid] = cosexc[m0 + tid];
  __syncthreads();

  // Per-lane address pieces for the async staging (EXEC is all-ones here).
  // Each b128 op copies 32 lanes x 16 B = 512 B (4 rows of a 16-col chunk).
  const uint64_t cbase =
      (uint64_t)(uintptr_t)centn +
      ((uint64_t)(uint32_t)((lane >> 3) * (NDIM * 4) + (lane & 7) * 16));
  const uint32_t lbase =
      (uint32_t)(uintptr_t)(&bstage[0][0][0]) + (uint32_t)(w * 2048 + lane * 16);

  // Stage chunk (tile n0, dims [32q, 32q+31]) into buffer `buf`.
  auto stage = [&](int n0, int q, int buf) {
    const uint64_t g0 = cbase + (uint64_t)n0 * (NDIM * 4) + (uint32_t)(q * 128);
    const uint32_t l0 = lbase + (uint32_t)(buf * (8 * 2048));
#pragma unroll
    for (int i = 0; i < 4; ++i) {
      const uint64_t ga = g0 + (uint32_t)(i * 2048);  // 4 rows per op
      const uint32_t la = l0 + (uint32_t)(i * 512);
      asm volatile("global_load_async_to_lds_b128 %0, %1, off"
                   :: "v"(la), "v"(ga) : "memory");
    }
  };

  // Preload all A fragments for this wave's 16 rows (K = 128 -> 32 frags).
  const v2f* ap = (const v2f*)(xn + (size_t)(m0 + lm) * NDIM + 2 * hi);
  v2f a[32];
#pragma unroll
  for (int kk = 0; kk < 32; ++kk) a[kk] = ap[2 * kk];

  // Diagonal replacement values: row m0 + r + 8*hi has g = r.
  float dv[8];
#pragma unroll
  for (int r = 0; r < 8; ++r) dv[r] = LSCALE * ce_s[r + 8 * hi];
  const int spkcol = 2 * blockIdx.x + hi;  // own-speaker column for this half

  float sums[8] = {0.f, 0.f, 0.f, 0.f, 0.f, 0.f, 0.f, 0.f};

  stage(w * 16, 0, 0);  // prologue: tile 0, chunk 0 -> buffer 0

  for (int j = 0; j < 16; ++j) {            // N-tiles owned by this wave
    const int n0 = w * 16 + 128 * j;
    v8f acc = {};
#pragma unroll
    for (int q = 0; q < 4; ++q) {           // K chunks of 32 dims
      const int st  = j * 4 + q;            // current pipeline step
      const int stn = st + 1;
      if (stn < 64) {
        stage(w * 16 + 128 * (stn >> 2), stn & 3, stn & 1);
        asm volatile("s_wait_asynccnt 0x4" ::: "memory");  // step st resident
      } else {
        asm volatile("s_wait_asynccnt 0x0" ::: "memory");
      }
      const float* cb = &bstage[st & 1][w][0] + lm * 32 + 2 * hi;
#pragma unroll
      for (int t = 0; t < 8; ++t) {
        const v2f b = *(const v2f*)(cb + 4 * t);
        acc = __builtin_amdgcn_wmma_f32_16x16x4_f32(
            false, a[q * 8 + t], false, b, (short)0, acc, false, false);
      }
    }
    const bool isdiag = (n0 + lm) == spkcol;
#pragma unroll
    for (int r = 0; r < 8; ++r) {
      float val = LSCALE * acc[r];
      if (isdiag) val = dv[r];
      val = fmaxf(val, 1e-6f);
      sums[r] += __expf(val);
    }
  }

  // Reduce across the 16 lanes of each half (halves are distinct rows).
#pragma unroll
  for (int off = 8; off >= 1; off >>= 1)
#pragma unroll
    for (int r = 0; r < 8; ++r) sums[r] += __shfl_xor(sums[r], off, 32);

  if (lm == 0) {
#pragma unroll
    for (int r = 0; r < 8; ++r) part[w][r + 8 * hi] = sums[r];
  }
  __syncthreads();

  if (tid < 16) {
    float tot = 0.f;
#pragma unroll
    for (int ww = 0; ww < 8; ++ww) tot += part[ww][tid];
    const float lse = __logf(tot);
    const float lab = fmaxf(LSCALE * ce_s[tid], 1e-6f);
    atomicAdd(out, (lse - lab) * (1.0f / (float)NROWS));
  }
}

__global__ void ge2e_zero(float* out) { out[0] = 0.0f; }

// ---------------------------------------------------------------------------
extern "C" void kernel_launch(void* const* d_in, const int* in_sizes, int n_in,
                              void* d_out, int out_size, void* d_ws,
                              size_t ws_size, hipStream_t stream) {
  (void)in_sizes; (void)n_in; (void)out_size; (void)ws_size;
  const float* x = (const float*)d_in[0];
  float* out = (float*)d_out;

  char* ws = (char*)d_ws;
  float* xn     = (float*)(ws);                                     // 8 MB
  float* centn  = (float*)(ws + (size_t)NROWS * NDIM * 4);          // 1 MB
  float* cosexc = (float*)(ws + (size_t)(NROWS + NSPK) * NDIM * 4); // 64 KB

  ge2e_zero<<<1, 1, 0, stream>>>(out);
  ge2e_phase1<<<NSPK, 256, 0, stream>>>(x, xn, centn, cosexc);
  ge2e_phase2<<<NROWS / 16, 256, 0, stream>>>(xn, centn, cosexc, out);
}